// SegmentScore_2087354106211
// MI455X (gfx1250) — compile-verified
//
#include <hip/hip_runtime.h>

#define B_   4
#define T_   256
#define L_   216
#define LP   224     // L padded to multiple of 16
#define KD   44      // fused feature dim (42) padded to multiple of 4
#define CSD  40      // prefix-sum feature dim (12 pc + 12 root + 16 qual)
#define W_NM  0.1f
#define W_ADV 0.001f

typedef __attribute__((ext_vector_type(2))) float v2f;
typedef __attribute__((ext_vector_type(8))) float v8f;

// ---- kernel 1: per-batch prefix sums over T of the 40-dim fused frame features
__global__ void seg_prefix_scan(const float* __restrict__ apc,
                                const float* __restrict__ aroot,
                                const float* __restrict__ aqual,
                                const float* __restrict__ iroot,
                                const float* __restrict__ iqual,
                                float* __restrict__ cs) {
  int b = blockIdx.x;
  int d = threadIdx.x;
  if (d >= CSD) return;
  float acc = 0.0f;
  cs[(b * (T_ + 1) + 0) * CSD + d] = 0.0f;
  for (int t = 0; t < T_; ++t) {
    int ft = b * T_ + t;
    float v;
    if (d < 12)      v = apc[ft * 12 + d];
    else if (d < 24) v = aroot[ft * 12 + (d - 12)] - W_ADV * iroot[ft * 12 + (d - 12)];
    else             v = aqual[ft * 16 + (d - 24)] - W_ADV * iqual[ft * 16 + (d - 24)];
    acc += v;
    cs[(b * (T_ + 1) + t + 1) * CSD + d] = acc;
  }
}

// ---- kernel 2: fused & padded harmony matrix G[b][LP][KD]
__global__ void seg_harmony(const float* __restrict__ hpc,
                            const float* __restrict__ hroot,
                            const float* __restrict__ hqual,
                            const int* __restrict__ pc_only,
                            float* __restrict__ G) {
  int b = blockIdx.x;
  int l = threadIdx.x;
  if (l >= LP) return;
  float* g = G + (size_t)(b * LP + l) * KD;
  if (l < L_) {
    float hs = 0.0f, hp[12];
#pragma unroll
    for (int d = 0; d < 12; ++d) { hp[d] = hpc[(b * L_ + l) * 12 + d]; hs += hp[d]; }
    float scale = (1.0f + 2.0f * W_NM) - W_ADV / hs;
#pragma unroll
    for (int d = 0; d < 12; ++d) g[d] = hp[d] * scale;
    g[12] = -W_NM;          // pairs with feature MpcSum
    g[13] = -W_NM * hs;     // pairs with constant-1 feature
    float rs = (pc_only[0] != 0) ? 0.0f : 1.0f;
#pragma unroll
    for (int d = 0; d < 12; ++d) g[14 + d] = rs * hroot[(b * L_ + l) * 12 + d];
#pragma unroll
    for (int d = 0; d < 16; ++d) g[26 + d] = rs * hqual[(b * L_ + l) * 16 + d];
    g[42] = 0.0f; g[43] = 0.0f;
  } else {
#pragma unroll
    for (int i = 0; i < KD; ++i) g[i] = 0.0f;
  }
}

// ---- kernel 3: per (b,s) block: build 256xKD feature slab in LDS, WMMA-GEMM
//      against G[b] (also in LDS), stream masked 16x16 fp32 tiles to HBM.
__global__ void __launch_bounds__(256)
seg_score_wmma(const float* __restrict__ cs,
               const float* __restrict__ G,
               float* __restrict__ out) {
  extern __shared__ float smem[];
  float* sG = smem;              // LP*KD floats
  float* sF = smem + LP * KD;    // T_*KD floats
  int b = blockIdx.x / T_;
  int s = blockIdx.x % T_;
  int tid = threadIdx.x;

  // stage fused harmony matrix for this batch
  for (int i = tid; i < LP * KD; i += 256) sG[i] = G[(size_t)b * LP * KD + i];

  // one thread per segment end e: build F[e][0..43]
  {
    int e = tid;
    int len = e - s + 1;
    float inv = 1.0f / (float)(len > 0 ? len : 1);   // e<s rows are masked at store
    const float* ce = cs + (b * (T_ + 1) + e + 1) * CSD;
    const float* c0 = cs + (b * (T_ + 1) + s) * CSD;
    float* f = sF + e * KD;
    float pcsum = 0.0f;
#pragma unroll
    for (int d = 0; d < 12; ++d) { float v = (ce[d] - c0[d]) * inv; f[d] = v; pcsum += v; }
    f[12] = pcsum;   // MpcSum
    f[13] = 1.0f;    // constant feature
#pragma unroll
    for (int d = 12; d < CSD; ++d) f[d + 2] = (ce[d] - c0[d]) * inv;
    f[42] = 0.0f; f[43] = 0.0f;
  }
  __syncthreads();

  int wave = tid >> 5, lane = tid & 31;
  int hi = lane >> 4, lo = lane & 15;   // f32 A/B fragment: lane row = lo, K base offset = 2*hi

  const int NT = (T_ / 16) * (LP / 16); // 16 e-tiles * 14 l-tiles = 224 tiles
  for (int tile = wave; tile < NT; tile += 8) {
    int et = tile / (LP / 16);
    int lt = tile % (LP / 16);
    const float* fr = sF + (et * 16 + lo) * KD + 2 * hi;
    const float* gr = sG + (lt * 16 + lo) * KD + 2 * hi;
    v8f acc = {};
#pragma unroll
    for (int k = 0; k < KD; k += 4) {
      v2f a  = *(const v2f*)(fr + k);   // K = k + 2*hi, k + 2*hi + 1
      v2f bb = *(const v2f*)(gr + k);
      acc = __builtin_amdgcn_wmma_f32_16x16x4_f32(false, a, false, bb,
                                                  (short)0, acc, false, false);
    }
    int lcol = lt * 16 + lo;
    if (lcol < L_) {
      float* op = out + (size_t)(b * T_ + s) * T_ * L_ + lcol;
#pragma unroll
      for (int r = 0; r < 8; ++r) {
        int e = et * 16 + r + 8 * hi;   // C/D layout: VGPR r -> row r + 8*(lane>=16)
        op[(size_t)e * L_] = (e >= s) ? acc[r] : 0.0f;
      }
    }
  }
}

extern "C" void kernel_launch(void* const* d_in, const int* in_sizes, int n_in,
                              void* d_out, int out_size, void* d_ws, size_t ws_size,
                              hipStream_t stream) {
  (void)in_sizes; (void)n_in; (void)out_size; (void)ws_size;
  const float* apc   = (const float*)d_in[0];
  const float* aroot = (const float*)d_in[1];
  const float* aqual = (const float*)d_in[2];
  // d_in[3] (inactive_pc) intentionally unused: reference uses active_pc there.
  const float* iroot = (const float*)d_in[4];
  const float* iqual = (const float*)d_in[5];
  const float* hpc   = (const float*)d_in[6];
  const float* hroot = (const float*)d_in[7];
  const float* hqual = (const float*)d_in[8];
  const int*   ponly = (const int*)d_in[9];
  float* out = (float*)d_out;

  float* cs = (float*)d_ws;                          // B*(T+1)*CSD floats
  float* G  = cs + (size_t)B_ * (T_ + 1) * CSD;      // B*LP*KD floats

  seg_prefix_scan<<<B_, 64, 0, stream>>>(apc, aroot, aqual, iroot, iqual, cs);
  seg_harmony<<<B_, 256, 0, stream>>>(hpc, hroot, hqual, ponly, G);

  size_t lds_bytes = (size_t)(LP * KD + T_ * KD) * sizeof(float);  // 84,480 B
  seg_score_wmma<<<B_ * T_, 256, lds_bytes, stream>>>(cs, G, out);
}